// MultiBoxLoss_55851754717734
// MI455X (gfx1250) — compile-verified
//
#include <hip/hip_runtime.h>
#include <math.h>

#ifndef USE_ASYNC_LDS
#define USE_ASYNC_LDS 1
#endif

#define MAX_M 64
#define BLOCK 256

typedef __attribute__((ext_vector_type(2))) float v2f;
typedef __attribute__((ext_vector_type(8))) float v8f;

// Full 32-lane sum using the CDNA5 matrix pipe.
// A: vgpr0 = v (A[m,0]=v[m], A[m,2]=v[m+16]), vgpr1 = 0. B = all-ones 4x16.
// D[m,n] = v[m] + v[m+16]. Lane n holds D[0..7,n] (n<16) / D[8..15,n-16] in
// its 8 D VGPRs, so sum-of-8 + xor-16 shuffle = total in every lane.
// Requires EXEC all ones (call only at full block/wave convergence).
__device__ __forceinline__ float wave_sum_wmma(float v) {
  v2f a; a[0] = v;    a[1] = 0.0f;
  v2f b; b[0] = 1.0f; b[1] = 1.0f;
  v8f c = {};
  v8f d = __builtin_amdgcn_wmma_f32_16x16x4_f32(false, a, false, b,
                                                (short)0, c, false, false);
  float s = ((d[0] + d[1]) + (d[2] + d[3])) + ((d[4] + d[5]) + (d[6] + d[7]));
  s += __shfl_xor(s, 16, 32);
  return s;
}

__device__ __forceinline__ float smooth_l1(float d) {
  const float beta = 1.0f / 9.0f;
  return (d <= beta) ? (4.5f * d * d) : (d - 0.5f * beta);
}

// Branchless focal term, fast log (v_log_f32): ~2-3 ulp, fine for a 15.7M-term sum.
__device__ __forceinline__ float focal_term(float p, bool is_pos_class) {
  p = fminf(fmaxf(p, 1e-4f), 1.0f - 1e-4f);
  const float s = is_pos_class ? p : (1.0f - p);        // log argument
  const float u = 1.0f - s;                             // weight base
  const float alpha = is_pos_class ? 0.25f : 0.75f;
  return -alpha * u * u * __logf(s);
}

__global__ void __launch_bounds__(BLOCK) focal_partial_kernel(
    const float* __restrict__ cls,   // [B,A,C]
    const float* __restrict__ reg,   // [B,A,4]
    const float* __restrict__ anc,   // [A,4] (y1,x1,y2,x2)
    const float* __restrict__ ann,   // [B,M,5] (x1,y1,x2,y2,label)
    float* __restrict__ partial,     // [B, nblk, 3]
    int A, int C, int M)
{
  const int b = blockIdx.y;
  const int i = blockIdx.x * BLOCK + threadIdx.x;
  __shared__ float s_raw[MAX_M * 5];
  __shared__ __align__(16) float s_box[MAX_M][8];  // x1,y1,x2,y2,area,label,-,-
  __shared__ float s_red[3][BLOCK / 32];

  // ---- stage annotations for this image into LDS (async-to-LDS path) ----
  const float* gann = ann + (size_t)b * M * 5;
  const int nelem = M * 5;
  for (int idx = threadIdx.x; idx < nelem; idx += BLOCK) {
#if USE_ASYNC_LDS
    unsigned lds_off = (unsigned)(unsigned long long)(&s_raw[idx]);
    const float* g = gann + idx;
    asm volatile("global_load_async_to_lds_b32 %0, %1, off"
                 :: "v"(lds_off), "v"(g) : "memory");
#else
    s_raw[idx] = gann[idx];
#endif
  }
#if USE_ASYNC_LDS
  asm volatile("s_wait_asynccnt 0" ::: "memory");
#endif
  __syncthreads();

  // ---- repack: 32B/box, precompute area (saves per-thread recompute) ----
  for (int m = threadIdx.x; m < M; m += BLOCK) {
    const float x1 = s_raw[m * 5 + 0];
    const float y1 = s_raw[m * 5 + 1];
    const float x2 = s_raw[m * 5 + 2];
    const float y2 = s_raw[m * 5 + 3];
    s_box[m][0] = x1; s_box[m][1] = y1; s_box[m][2] = x2; s_box[m][3] = y2;
    s_box[m][4] = (x2 - x1) * (y2 - y1);
    s_box[m][5] = s_raw[m * 5 + 4];
  }
  __syncthreads();

  float cls_sum = 0.0f, reg_sum = 0.0f, pos_cnt = 0.0f;

  if (i < A) {
    const float4 av = *(const float4*)(anc + (size_t)i * 4);
    const float a0 = av.x;   // y1
    const float a1 = av.y;   // x1
    const float a2 = av.z;   // y2
    const float a3 = av.w;   // x2
    const float aw = a3 - a1, ah = a2 - a0;
    const float acx = a1 + 0.5f * aw, acy = a0 + 0.5f * ah;
    const float areaA = aw * ah;

    // ---- IoU max / argmax over GT boxes (first-max like jnp.argmax) ----
    float best = -1.0f; int bm = 0;
    for (int m = 0; m < M; ++m) {
      const float4 bb = *(const float4*)&s_box[m][0];   // broadcast b128
      const float areaB = s_box[m][4];
      float iw = fminf(a3, bb.z) - fmaxf(a1, bb.x);
      float ih = fminf(a2, bb.w) - fmaxf(a0, bb.y);
      iw = fmaxf(iw, 0.0f); ih = fmaxf(ih, 0.0f);
      const float inter = iw * ih;
      const float ua = fmaxf(areaA + areaB - inter, 1e-8f);
      const float iou = inter / ua;
      if (iou > best) { best = iou; bm = m; }
    }
    const bool positive = best >= 0.5f;
    const bool negative = best < 0.4f;

    // ---- classification focal loss (ignored anchors contribute 0) ----
    if (positive || negative) {
      const int label = positive ? (int)s_box[bm][5] : -1;
      const float* cp = cls + ((size_t)b * A + i) * C;
      if ((C & 3) == 0) {
        const float4* cp4 = (const float4*)cp;
        for (int c4 = 0; c4 < (C >> 2); ++c4) {
          float4 p4 = cp4[c4];
          const int cb = c4 << 2;
          cls_sum += focal_term(p4.x, cb + 0 == label);
          cls_sum += focal_term(p4.y, cb + 1 == label);
          cls_sum += focal_term(p4.z, cb + 2 == label);
          cls_sum += focal_term(p4.w, cb + 3 == label);
        }
      } else {
        for (int ci = 0; ci < C; ++ci)
          cls_sum += focal_term(cp[ci], ci == label);
      }
    }

    // ---- regression smooth-L1 for positives ----
    if (positive) {
      pos_cnt = 1.0f;
      const float4 gb = *(const float4*)&s_box[bm][0];
      const float gw  = fmaxf(gb.z - gb.x, 1.0f);
      const float gh  = fmaxf(gb.w - gb.y, 1.0f);
      const float gcx = gb.x + 0.5f * (gb.z - gb.x);
      const float gcy = gb.y + 0.5f * (gb.w - gb.y);
      const float4 r = *(const float4*)(reg + ((size_t)b * A + i) * 4);
      const float t0 = (gcy - acy) / ah;
      const float t1 = (gcx - acx) / aw;
      const float t2 = logf(gh / ah);
      const float t3 = logf(gw / aw);
      reg_sum = smooth_l1(fabsf(t0 - r.x)) + smooth_l1(fabsf(t1 - r.y)) +
                smooth_l1(fabsf(t2 - r.z)) + smooth_l1(fabsf(t3 - r.w));
    }
  }

  // ---- block reduction: WMMA wave-sum, then LDS across 8 waves ----
  const float wc = wave_sum_wmma(cls_sum);
  const float wr = wave_sum_wmma(reg_sum);
  const float wp = wave_sum_wmma(pos_cnt);
  const int wave = threadIdx.x >> 5;
  const int lane = threadIdx.x & 31;
  if (lane == 0) { s_red[0][wave] = wc; s_red[1][wave] = wr; s_red[2][wave] = wp; }
  __syncthreads();
  if (threadIdx.x == 0) {
    float tc = 0.0f, tr = 0.0f, tp = 0.0f;
    for (int w = 0; w < BLOCK / 32; ++w) {
      tc += s_red[0][w]; tr += s_red[1][w]; tp += s_red[2][w];
    }
    float* p = partial + ((size_t)b * gridDim.x + blockIdx.x) * 3;
    p[0] = tc; p[1] = tr; p[2] = tp;
  }
}

__global__ void __launch_bounds__(32) focal_finalize_kernel(
    const float* __restrict__ partial,  // [B, nblk, 3]
    float* __restrict__ out, int B, int nblk)
{
  const int lane = threadIdx.x & 31;
  float vc = 0.0f, vr = 0.0f;
  if (lane < B) {
    float cs = 0.0f, rs = 0.0f, np = 0.0f;
    const float* p = partial + (size_t)lane * nblk * 3;
    for (int k = 0; k < nblk; ++k) {
      cs += p[k * 3 + 0]; rs += p[k * 3 + 1]; np += p[k * 3 + 2];
    }
    vc = cs / fmaxf(np, 1.0f);
    const float r = rs / fmaxf(np * 4.0f, 1.0f);
    vr = (np > 0.0f) ? r : 0.0f;
  }
  // reconverged: full-EXEC WMMA reductions over images
  const float tc = wave_sum_wmma(vc);
  const float tr = wave_sum_wmma(vr);
  if (lane == 0) {
    out[0] = tc / (float)B;
    out[1] = tr / (float)B;
  }
}

extern "C" void kernel_launch(void* const* d_in, const int* in_sizes, int n_in,
                              void* d_out, int out_size, void* d_ws, size_t ws_size,
                              hipStream_t stream) {
  const float* cls = (const float*)d_in[0];
  const float* reg = (const float*)d_in[1];
  const float* anc = (const float*)d_in[2];
  const float* ann = (const float*)d_in[3];
  float* out = (float*)d_out;
  float* partial = (float*)d_ws;

  const int A = in_sizes[2] / 4;             // anchors: [1,A,4]
  const int B = in_sizes[1] / (A * 4);       // regressions: [B,A,4]
  const int C = in_sizes[0] / ((size_t)B * A);
  int M = in_sizes[3] / (B * 5);
  if (M > MAX_M) M = MAX_M;

  const int nblk = (A + BLOCK - 1) / BLOCK;
  dim3 grid(nblk, B);
  focal_partial_kernel<<<grid, BLOCK, 0, stream>>>(cls, reg, anc, ann,
                                                   partial, A, C, M);
  focal_finalize_kernel<<<1, 32, 0, stream>>>(partial, out, B, nblk);
}